// PolarizingBlock_39187281609230
// MI455X (gfx1250) — compile-verified
//
#include <hip/hip_runtime.h>

typedef float v2f __attribute__((ext_vector_type(2)));
typedef float v8f __attribute__((ext_vector_type(8)));
typedef float f4  __attribute__((ext_vector_type(4)));

#define B_   16
#define T_   4096
#define D_   1024
#define H_   32
#define TCH  4                    // T-chunks per (b, d-tile) for occupancy
#define TPER (T_ / TCH)           // 1024 t per wave
#define DT   (D_ / 16)            // 64 d-tiles of width 16
#define NWAVES (B_ * DT * TCH)    // 4096 waves

// workspace layout (floats)
#define WS_ACC_R 0
#define WS_ACC_I (B_ * D_)
#define WS_CNT   (2 * B_ * D_)
#define WS_ANR   (2 * B_ * D_ + 32)
#define WS_ANI   (3 * B_ * D_ + 32)

__device__ __forceinline__ float gelu_exact(float x) {
    return 0.5f * x * (1.0f + erff(x * 0.70710678118654752440f));
}

// ---------------- kernel 1: zero atomic accumulators ----------------
__global__ void zero_acc(float* ws) {
    int i = blockIdx.x * blockDim.x + threadIdx.x;
    if (i < 2 * B_ * D_) ws[i] = 0.0f;
}

// ---------------- kernel 2: count[b] = max(sum_t mask, 1) ----------------
__global__ void count_kernel(const float* __restrict__ mask, float* __restrict__ cnt) {
    __shared__ float sm[256];
    const int b = blockIdx.x;
    float s = 0.0f;
    for (int t = threadIdx.x; t < T_; t += 256) s += mask[b * T_ + t];
    sm[threadIdx.x] = s;
    __syncthreads();
    for (int off = 128; off > 0; off >>= 1) {
        if (threadIdx.x < off) sm[threadIdx.x] += sm[threadIdx.x + off];
        __syncthreads();
    }
    if (threadIdx.x == 0) cnt[b] = fmaxf(sm[0], 1.0f);
}

// ---------------- kernel 3: masked T-reduction via V_WMMA_F32_16X16X4_F32 ----
// A(16x4) = mask chunk broadcast to all 16 rows; B(4x16) = Z tile (4 t x 16 d).
// All 16 rows of the accumulator hold identical column sums; row 0 is merged.
__global__ void reduce_wmma(const float* __restrict__ Zr, const float* __restrict__ Zi,
                            const float* __restrict__ mask,
                            float* __restrict__ acc_r, float* __restrict__ acc_i) {
    const int lane = threadIdx.x & 31;
    const int w    = blockIdx.x * (blockDim.x >> 5) + (threadIdx.x >> 5);
    const int tch  = w & (TCH - 1);
    const int dt   = (w / TCH) & (DT - 1);
    const int b    = w / (TCH * DT);
    const int d0   = dt * 16;
    const int t0   = tch * TPER;

    // ISA layout: VGPR0 carries K={0,2} (lanes 0-15 / 16-31), VGPR1 carries K={1,3}
    const int row = (lane < 16) ? 0 : 2;
    const int col = lane & 15;

    const float* pzr = Zr + ((size_t)b * T_ + (size_t)(t0 + row)) * D_ + d0 + col;
    const float* pzi = Zi + ((size_t)b * T_ + (size_t)(t0 + row)) * D_ + d0 + col;
    const float* pm  = mask + (size_t)b * T_ + t0 + row;

    v8f accR = {};
    v8f accI = {};
    for (int it = 0; it < TPER / 4; ++it) {
        v2f a;
        a.x = pm[0];                                   // mask[t0+row]   (K=0/2)
        a.y = pm[1];                                   // mask[t0+row+1] (K=1/3)
        v2f br, bi;
        br.x = __builtin_nontemporal_load(pzr);        // Z[t0+row  ][d0+col]
        br.y = __builtin_nontemporal_load(pzr + D_);   // Z[t0+row+1][d0+col]
        bi.x = __builtin_nontemporal_load(pzi);
        bi.y = __builtin_nontemporal_load(pzi + D_);
        accR = __builtin_amdgcn_wmma_f32_16x16x4_f32(false, a, false, br, (short)0, accR, false, false);
        accI = __builtin_amdgcn_wmma_f32_16x16x4_f32(false, a, false, bi, (short)0, accI, false, false);
        pm  += 4;
        pzr += 4 * D_;
        pzi += 4 * D_;
    }
    // C/D layout: VGPR0 lanes 0-15 = (M=0, N=lane). All rows equal -> use row 0 only.
    if (lane < 16) {
        atomicAdd(&acc_r[b * D_ + d0 + lane], accR[0]);
        atomicAdd(&acc_i[b * D_ + d0 + lane], accI[0]);
    }
}

// ---------------- kernel 4: tiny per-(b,d) MLPs ----------------
__global__ void mlp_kernel(const float* __restrict__ acc_r, const float* __restrict__ acc_i,
                           const float* __restrict__ cnt,
                           const float* __restrict__ W1m, const float* __restrict__ b1m,
                           const float* __restrict__ W2m, const float* __restrict__ b2m,
                           const float* __restrict__ W1p, const float* __restrict__ b1p,
                           const float* __restrict__ W2p, const float* __restrict__ b2p,
                           const float* __restrict__ mag_scale_p,
                           float* __restrict__ anr, float* __restrict__ ani) {
    const int idx = blockIdx.x * blockDim.x + threadIdx.x;
    if (idx >= B_ * D_) return;
    const int b = idx / D_;

    const float c  = cnt[b];
    const float Ar = acc_r[idx] / c;
    const float Ai = acc_i[idx] / c;

    const float mag     = sqrtf(Ar * Ar + Ai * Ai);
    const float log_mag = logf(mag + 1e-6f);
    const float p0      = Ar / (mag + 1e-6f);
    const float p1      = Ai / (mag + 1e-6f);

    // magnitude MLP: 1 -> H -> 1
    float mag_delta = b2m[0];
#pragma unroll
    for (int h = 0; h < H_; ++h) {
        float t = log_mag * W1m[h] + b1m[h];
        mag_delta += gelu_exact(t) * W2m[h];
    }
    const float log_mag_out = log_mag + mag_scale_p[0] * mag_delta;

    // phase MLP: 2 -> H -> 2
    float pv0 = b2p[0];
    float pv1 = b2p[1];
#pragma unroll
    for (int h = 0; h < H_; ++h) {
        float t = p0 * W1p[2 * h] + p1 * W1p[2 * h + 1] + b1p[h];
        float g = gelu_exact(t);
        pv0 += g * W2p[h];
        pv1 += g * W2p[H_ + h];
    }
    const float nrm = fmaxf(sqrtf(pv0 * pv0 + pv1 * pv1), 1e-12f);
    pv0 /= nrm;
    pv1 /= nrm;

    const float r = expf(log_mag_out);
    anr[idx] = r * pv0;
    ani[idx] = r * pv1;
}

// ---------------- kernel 5: streaming broadcast-add (non-temporal) ----------
__global__ void apply_kernel(const float* __restrict__ Zr, const float* __restrict__ Zi,
                             const float* __restrict__ anr, const float* __restrict__ ani,
                             float* __restrict__ out) {
    const long long N4 = (long long)B_ * T_ * D_ / 4;
    const long long i  = (long long)blockIdx.x * blockDim.x + threadIdx.x;
    if (i >= N4) return;

    const int d4pb = D_ / 4;            // 256 float4 per row
    const int td4  = T_ * d4pb;         // float4 per batch
    const int b    = (int)(i / td4);
    const int d4   = (int)(i % d4pb);

    f4 ar = ((const f4*)anr)[b * d4pb + d4];   // reused -> regular (cached) load
    f4 ai = ((const f4*)ani)[b * d4pb + d4];
    f4 zr = __builtin_nontemporal_load(((const f4*)Zr) + i);
    f4 zi = __builtin_nontemporal_load(((const f4*)Zi) + i);

    f4* o = (f4*)out;
    __builtin_nontemporal_store(zr + ar, o + i);        // out[0]
    __builtin_nontemporal_store(zi + ai, o + N4 + i);   // out[1]
}

extern "C" void kernel_launch(void* const* d_in, const int* in_sizes, int n_in,
                              void* d_out, int out_size, void* d_ws, size_t ws_size,
                              hipStream_t stream) {
    const float* Zr   = (const float*)d_in[0];
    const float* Zi   = (const float*)d_in[1];
    const float* mask = (const float*)d_in[2];
    const float* W1m  = (const float*)d_in[3];
    const float* b1m  = (const float*)d_in[4];
    const float* W2m  = (const float*)d_in[5];
    const float* b2m  = (const float*)d_in[6];
    const float* W1p  = (const float*)d_in[7];
    const float* b1p  = (const float*)d_in[8];
    const float* W2p  = (const float*)d_in[9];
    const float* b2p  = (const float*)d_in[10];
    const float* msc  = (const float*)d_in[11];

    float* ws    = (float*)d_ws;
    float* acc_r = ws + WS_ACC_R;
    float* acc_i = ws + WS_ACC_I;
    float* cnt   = ws + WS_CNT;
    float* anr   = ws + WS_ANR;
    float* ani   = ws + WS_ANI;
    float* out   = (float*)d_out;

    // 1) zero accumulators (ws is poisoned / left dirty between calls)
    zero_acc<<<(2 * B_ * D_ + 255) / 256, 256, 0, stream>>>(ws);

    // 2) per-batch mask counts
    count_kernel<<<B_, 256, 0, stream>>>(mask, cnt);

    // 3) masked T-reduction on the matrix pipe (f32 WMMA)
    reduce_wmma<<<NWAVES / 8, 256, 0, stream>>>(Zr, Zi, mask, acc_r, acc_i);

    // 4) tiny MLPs on the 16384 reduced values
    mlp_kernel<<<(B_ * D_ + 255) / 256, 256, 0, stream>>>(
        acc_r, acc_i, cnt, W1m, b1m, W2m, b2m, W1p, b1p, W2p, b2p, msc, anr, ani);

    // 5) streaming broadcast-add into the (2,B,T,D) output
    const long long N4 = (long long)B_ * T_ * D_ / 4;
    apply_kernel<<<(unsigned)((N4 + 255) / 256), 256, 0, stream>>>(Zr, Zi, anr, ani, out);
}